// BSplineFFD_73057393705597
// MI455X (gfx1250) — compile-verified
//
#include <hip/hip_runtime.h>
#include <math.h>
#include <stdint.h>

// 16-byte vector with 4-byte alignment: lets us issue global_load_b128 on
// addresses that are only dword-aligned (row offsets are multiples of 12 B),
// legal on CDNA5 (DWORD alignment mode aligns to min(elem,4)=4 B).
typedef float v4f __attribute__((ext_vector_type(4), aligned(4)));

#define BLOCK 256

__device__ __forceinline__ void bspline_w(float u, float* w) {
    const float s  = 1.0f / 6.0f;
    const float u2 = u * u;
    const float u3 = u2 * u;
    w[0] = (1.0f - 3.0f * u + 3.0f * u2 - u3) * s;
    w[1] = (4.0f - 6.0f * u2 + 3.0f * u3) * s;
    w[2] = (1.0f + 3.0f * u + 3.0f * u2 - 3.0f * u3) * s;
    w[3] = u3 * s;
}

__global__ __launch_bounds__(BLOCK) void BSplineFFD_kernel(
    const float* __restrict__ verts,
    const float* __restrict__ dG,
    const float* __restrict__ origin,
    const float* __restrict__ spacing,
    float* __restrict__ out,
    int n, int nx, int ny, int nz)
{
    const int tid        = threadIdx.x;
    const int block_base = blockIdx.x * BLOCK;   // first vertex of this block

    // ---- Stage the block's vertex slab (3 KB) into LDS with the CDNA5
    // ---- async global->LDS path (ASYNCcnt). 768 dwords, 256 lanes x 3.
    __shared__ float sverts[BLOCK * 3];
    const int base_dw = block_base * 3;
    const int n_dw    = n * 3;
#pragma unroll
    for (int r = 0; r < 3; ++r) {
        const int dw  = tid + r * BLOCK;         // dword within slab [0,768)
        const int gdw = base_dw + dw;
        if (gdw < n_dw) {                        // per-lane EXEC guard (ragged tail)
            const uint32_t lds_off = (uint32_t)(uintptr_t)(&sverts[dw]);
            const float*   gp      = verts + gdw;
            asm volatile("global_load_async_to_lds_b32 %0, %1, off"
                         :: "v"(lds_off), "v"(gp) : "memory");
        }
    }
    asm volatile("s_wait_asynccnt 0x0" ::: "memory");  // this wave's DMAs done
    __syncthreads();                                    // all waves' DMAs visible

    const int i = block_base + tid;
    if (i >= n) return;

    const float vx = sverts[3 * tid + 0];
    const float vy = sverts[3 * tid + 1];
    const float vz = sverts[3 * tid + 2];

    const float rx = (vx - origin[0]) / spacing[0];
    const float ry = (vy - origin[1]) / spacing[1];
    const float rz = (vz - origin[2]) / spacing[2];

    const float fx = floorf(rx), fy = floorf(ry), fz = floorf(rz);
    const float ux = fminf(fmaxf(rx - fx, 0.0f), 1.0f);
    const float uy = fminf(fmaxf(ry - fy, 0.0f), 1.0f);
    const float uz = fminf(fmaxf(rz - fz, 0.0f), 1.0f);

    float wa[4], wb[4], wc[4];
    bspline_w(ux, wa);
    bspline_w(uy, wb);
    bspline_w(uz, wc);

    const int ia0 = (int)fx - 1;
    const int ib0 = (int)fy - 1;
    const int ic0 = (int)fz - 1;

    // Boundary masking == zero the per-axis weight (product weight becomes 0,
    // matching the reference's where(mask, w, 0)); clamp the index so the
    // gather address stays in-bounds (gathered value is multiplied by 0).
    int ja[4], jb[4], jc[4];
#pragma unroll
    for (int t = 0; t < 4; ++t) {
        int a = ia0 + t;
        if (a < 0 || a >= nx) wa[t] = 0.0f;
        ja[t] = a < 0 ? 0 : (a >= nx ? nx - 1 : a);
        int b = ib0 + t;
        if (b < 0 || b >= ny) wb[t] = 0.0f;
        jb[t] = b < 0 ? 0 : (b >= ny ? ny - 1 : b);
        int c = ic0 + t;
        if (c < 0 || c >= nz) wc[t] = 0.0f;
        jc[t] = c < 0 ? 0 : (c >= nz ? nz - 1 : c);
    }

    // Warm WGP$ for all 16 neighborhood rows (gfx1250 global_prefetch_b8)
    // before the dependent FMA chain below.
#pragma unroll
    for (int a = 0; a < 4; ++a) {
#pragma unroll
        for (int b = 0; b < 4; ++b) {
            const int rowflat = ((ja[a] * ny + jb[b]) * nz + jc[0]) * 3;
            __builtin_prefetch(dG + rowflat, 0, 3);
        }
    }

    const bool zin = (ic0 >= 0) && (ic0 + 3 < nz);  // all 4 z-points contiguous & valid

    float ax = 0.0f, ay = 0.0f, az = 0.0f;
#pragma unroll
    for (int a = 0; a < 4; ++a) {
#pragma unroll
        for (int b = 0; b < 4; ++b) {
            const float wab = wa[a] * wb[b];
            const int   rowbase = (ja[a] * ny + jb[b]) * nz;  // < 96^3, int math OK
            if (zin) {
                // 4 consecutive z-points = 48 contiguous bytes: 3x b128 loads.
                const float* p  = dG + (rowbase + ic0) * 3;
                const v4f    g0 = *(const v4f*)(p + 0);  // c0.x c0.y c0.z c1.x
                const v4f    g1 = *(const v4f*)(p + 4);  // c1.y c1.z c2.x c2.y
                const v4f    g2 = *(const v4f*)(p + 8);  // c2.z c3.x c3.y c3.z
                const float w0 = wab * wc[0], w1 = wab * wc[1];
                const float w2 = wab * wc[2], w3 = wab * wc[3];
                ax = fmaf(w0, g0.x, ax); ay = fmaf(w0, g0.y, ay); az = fmaf(w0, g0.z, az);
                ax = fmaf(w1, g0.w, ax); ay = fmaf(w1, g1.x, ay); az = fmaf(w1, g1.y, az);
                ax = fmaf(w2, g1.z, ax); ay = fmaf(w2, g1.w, ay); az = fmaf(w2, g2.x, az);
                ax = fmaf(w3, g2.y, ax); ay = fmaf(w3, g2.z, ay); az = fmaf(w3, g2.w, az);
            } else {
                // Boundary lanes (~9%): per-point gather with clamped indices.
#pragma unroll
                for (int c = 0; c < 4; ++c) {
                    const float  w = wab * wc[c];
                    const float* q = dG + (rowbase + jc[c]) * 3;
                    ax = fmaf(w, q[0], ax);
                    ay = fmaf(w, q[1], ay);
                    az = fmaf(w, q[2], az);
                }
            }
        }
    }

    // Streamed once: non-temporal stores so out doesn't evict L2-resident deltaG.
    __builtin_nontemporal_store(vx + ax, &out[3 * i + 0]);
    __builtin_nontemporal_store(vy + ay, &out[3 * i + 1]);
    __builtin_nontemporal_store(vz + az, &out[3 * i + 2]);
}

extern "C" void kernel_launch(void* const* d_in, const int* in_sizes, int n_in,
                              void* d_out, int out_size, void* d_ws, size_t ws_size,
                              hipStream_t stream) {
    const float* verts   = (const float*)d_in[0];
    const float* dG      = (const float*)d_in[1];
    const float* origin  = (const float*)d_in[2];
    const float* spacing = (const float*)d_in[3];
    float*       out     = (float*)d_out;

    const int n = in_sizes[0] / 3;

    // Recover the (cubic) grid edge length from deltaG's flat size = g^3 * 3.
    const long long cells = (long long)in_sizes[1] / 3;
    int g = (int)llround(cbrt((double)cells));
    while ((long long)g * g * g < cells) ++g;
    while (g > 1 && (long long)g * g * g > cells) --g;

    const int grid = (n + BLOCK - 1) / BLOCK;
    BSplineFFD_kernel<<<grid, BLOCK, 0, stream>>>(verts, dG, origin, spacing,
                                                  out, n, g, g, g);
}